// Prototype_6141803233729
// MI455X (gfx1250) — compile-verified
//
#include <hip/hip_runtime.h>
#include <hip/hip_bf16.h>

// ---- problem constants ------------------------------------------------------
#define B_  2
#define S_  208
#define G_  84
#define V_  48
#define VOL 4096          // 16*16*16
#define TAPS 125          // 5*5*5
#define K_REAL 250        // Hr taps + Hi taps
#define K_PAD  252        // padded to multiple of 4
#define KSTEPS 63         // K_PAD / 4
#define MTILES 6          // 96 channels (48 re + 48 im) / 16
#define APACK_FLOATS (KSTEPS * MTILES * 32 * 2)   // 24192 floats, fragment-packed A
#define H_FLOATS (B_ * G_ * VOL)                  // 688128 per component
#define IM_OFFSET ((size_t)B_ * G_ * V_ * VOL)    // 33030144 floats
#define MSPLIT 4                                  // voxel-slabs per volume
// padded halo volume: 20x20x20 per component (z,y pitch 400/20)
#define PDIM 20
#define PVOL (PDIM * PDIM * PDIM)                 // 8000
#define HPAD_FLOATS (2 * PVOL)                    // 16000 (Hr @0, Hi @8000)
#define KTAB_INTS 256                             // 252 used, padded
#define CONV_LDS_FLOATS (APACK_FLOATS + HPAD_FLOATS + KTAB_INTS)  // 40448 -> 161792 B
#define S1_TILEVOX 32
#define S1_LDS_FLOATS (S_ * S1_TILEVOX * 2 + S_)  // x tile + so3 weights

typedef float v2f __attribute__((ext_vector_type(2)));
typedef float v8f __attribute__((ext_vector_type(8)));
typedef int   v2i __attribute__((ext_vector_type(2)));

// ---- kernel 0: pre-pack complex conv kernels into WMMA A-fragment layout ----
// A[ch][k], ch 0..95 (0..47 = re rows [kr | -ki], 48..95 = im rows [ki | kr]),
// k 0..251 (>=250 zero pad -> lets B-side skip masking entirely).
// Packed [kk][mt][lane][2] so the 6 per-K-step ds_load_b64 share one base with
// immediate offsets mt*256B:  idx = ((kk*6 + mt)*32 + lane)*2 + j
// ch = mt*16 + (lane&15),  k = kk*4 + (lane>>4)*2 + j   (ISA 7.12.2, 32-bit A)
__global__ void __launch_bounds__(256) prepack_kernel(
    const float* __restrict__ kr, const float* __restrict__ ki,
    float* __restrict__ apack)
{
    int idx = blockIdx.x * 256 + threadIdx.x;
    if (idx >= APACK_FLOATS) return;
    int j    = idx & 1;
    int lane = (idx >> 1) & 31;
    int pair = idx >> 6;
    int mt   = pair % MTILES;
    int kk   = pair / MTILES;
    int ch   = mt * 16 + (lane & 15);
    int k    = kk * 4 + (lane >> 4) * 2 + j;
    float val = 0.0f;
    if (k < TAPS) {
        val = (ch < V_) ? kr[ch * TAPS + k] : ki[(ch - V_) * TAPS + k];
    } else if (k < K_REAL) {
        int t = k - TAPS;
        val = (ch < V_) ? -ki[ch * TAPS + t] : kr[(ch - V_) * TAPS + t];
    }
    apack[idx] = val;
}

// ---- kernel 1: S-contraction  H[b,g,vox] = sum_s x * (cache*w) --------------
__global__ void __launch_bounds__(256) stage1_kernel(
    const float* __restrict__ img,   // (B,1,S,16,16,16,2)
    const float* __restrict__ cache, // (G,S,2)
    const float* __restrict__ so3,   // (S)
    float* __restrict__ Hr, float* __restrict__ Hi)
{
    extern __shared__ float s1[];
    float* xs = s1;                       // [S][32 vox][2]
    float* sw = s1 + S_ * S1_TILEVOX * 2; // [S]

    int tid = threadIdx.x;
    int b   = blockIdx.x >> 7;            // 128 tiles per b
    int vb  = (blockIdx.x & 127) * S1_TILEVOX;

    const float* src = img + (size_t)b * S_ * VOL * 2;
    for (int i = tid; i < S_ * S1_TILEVOX * 2; i += 256) {
        int s = i >> 6;                   // 64 floats per s (32 vox * 2)
        int j = i & 63;
        // 13.6 MB read exactly once across the grid -> non-temporal load
        xs[i] = __builtin_nontemporal_load(&src[((size_t)s * VOL + vb) * 2 + j]);
    }
    for (int i = tid; i < S_; i += 256) sw[i] = so3[i];
    __syncthreads();

    int wave = tid >> 5, lane = tid & 31;
    for (int g = wave; g < G_; g += 8) {
        const float* cg = cache + g * S_ * 2;
        float ar = 0.0f, ai = 0.0f;
        for (int s = 0; s < S_; ++s) {
            float w  = sw[s];
            float cr = cg[2 * s] * w;
            float ci = cg[2 * s + 1] * w;
            float xr = xs[(s * S1_TILEVOX + lane) * 2];
            float xi = xs[(s * S1_TILEVOX + lane) * 2 + 1];
            ar = fmaf(xr, cr, fmaf(xi, ci, ar));
            ai = fmaf(xi, cr, fmaf(-xr, ci, ai));
        }
        size_t o = (size_t)(b * G_ + g) * VOL + vb + lane;
        Hr[o] = ar;     // H stays L2-resident (read 4x by conv kernel)
        Hi[o] = ai;
    }
}

// ---- kernel 2: complex 5^3 conv as WMMA GEMM --------------------------------
// Per volume (b,g): out[ch 0..95][vox] = sum_k A[ch][k]*B[k][vox].
// H staged into LDS with a zero halo (20^3 per component) so the hot loop's
// im2col is just  sH[voxpad + ktab[k]]  -- no bounds checks, no div/mod.
// Each wave register-blocks TWO voxel tiles: every A fragment feeds 2 WMMAs,
// => 11 DS ops per 12 WMMAs in the K-loop (halves LDS bytes per FLOP).
// Output (264 MB, write-once stream) stored non-temporally to preserve L2
// for the A-pack + H staging data that all 672 blocks re-read.
__global__ void __launch_bounds__(256) conv_wmma_kernel(
    const float* __restrict__ apack,
    const float* __restrict__ Hr, const float* __restrict__ Hi,
    float* __restrict__ out)
{
    extern __shared__ float lds[];
    float* sA   = lds;                          // 24192 floats
    float* sH   = lds + APACK_FLOATS;           // 16000 floats (Hr@0, Hi@8000)
    int*   ktab = (int*)(lds + APACK_FLOATS + HPAD_FLOATS);  // 252 ints

    int tid  = threadIdx.x;
    int bg   = blockIdx.x / MSPLIT;             // 0..167  (b*G+g)
    int part = blockIdx.x % MSPLIT;

    // --- zero padded H, load A-pack, build per-k offset table ---
    for (int i = tid; i < PVOL; i += 256)       // 8000 float2 = 16000 floats
        *(v2f*)(sH + i * 2) = (v2f){0.f, 0.f};
    for (int i = tid; i < APACK_FLOATS; i += 256) sA[i] = apack[i];
    if (tid < K_PAD) {
        int k     = tid;
        int isHi  = (k >= TAPS) ? 1 : 0;
        int tap   = isHi ? (k - TAPS) : k;
        int delta = 0;                          // k>=250: A rows are zero, any addr ok
        if (k < K_REAL) {
            int dz = tap / 25, tr = tap % 25;
            int dy = tr / 5,  dx = tr % 5;
            delta = isHi * PVOL + (dz - 2) * (PDIM * PDIM) + (dy - 2) * PDIM + (dx - 2);
        }
        ktab[k] = delta;
    }
    __syncthreads();

    // --- fill interior of padded H (float2 row chunks; rows start at 8B align)
    const float* hrg = Hr + (size_t)bg * VOL;
    const float* hig = Hi + (size_t)bg * VOL;
    for (int t = tid; t < 2 * 256 * 8; t += 256) {   // 2 comps * 256 rows * 8 chunks
        int comp = t >> 11;
        int rc   = t & 2047;
        int r    = rc >> 3;          // row: z*16 + y
        int c    = rc & 7;           // 2-float chunk in x
        int z = r >> 4, y = r & 15;
        const float* srcp = (comp ? hig : hrg) + r * 16 + c * 2;
        float* dstp = sH + comp * PVOL + (z + 2) * (PDIM * PDIM) + (y + 2) * PDIM + 2 + c * 2;
        *(v2f*)dstp = *(const v2f*)srcp;
    }
    __syncthreads();

    int wave = tid >> 5, lane = tid & 31;
    int lhi  = lane >> 4;              // lane half selects K offset
    int llo  = lane & 15;              // N (voxel) within tile

    const int QT = 256 / MSPLIT;       // n-tiles per block (64)
    for (int qq = wave; qq < QT; qq += 16) {       // pair (qq, qq+8)
        int ntile0 = part * QT + qq;
        int ntile1 = ntile0 + 8;
        int vox0 = ntile0 * 16 + llo;
        int vox1 = ntile1 * 16 + llo;
        int z0 = vox0 >> 8, y0 = (vox0 >> 4) & 15, x0 = vox0 & 15;
        int z1 = vox1 >> 8, y1 = (vox1 >> 4) & 15, x1 = vox1 & 15;
        int vp0 = (z0 + 2) * (PDIM * PDIM) + (y0 + 2) * PDIM + (x0 + 2);
        int vp1 = (z1 + 2) * (PDIM * PDIM) + (y1 + 2) * PDIM + (x1 + 2);

        v8f acc0[MTILES], acc1[MTILES];
        #pragma unroll
        for (int mt = 0; mt < MTILES; ++mt) {
            v8f zf = {0.f, 0.f, 0.f, 0.f, 0.f, 0.f, 0.f, 0.f};
            acc0[mt] = zf;
            acc1[mt] = zf;
        }

        for (int kk = 0; kk < KSTEPS; ++kk) {
            // two consecutive table entries for this lane-half: k = 4kk+2*lhi+{0,1}
            v2i dd = *(const v2i*)(ktab + kk * 4 + lhi * 2);
            v2f bf0 = { sH[vp0 + dd.x], sH[vp0 + dd.y] };
            v2f bf1 = { sH[vp1 + dd.x], sH[vp1 + dd.y] };
            const float* abase = sA + (size_t)(kk * MTILES) * 64 + lane * 2;
            #pragma unroll
            for (int mt = 0; mt < MTILES; ++mt) {
                v2f afrag = *(const v2f*)(abase + mt * 64);
                acc0[mt] = __builtin_amdgcn_wmma_f32_16x16x4_f32(
                    false, afrag, false, bf0, (short)0, acc0[mt], false, false);
                acc1[mt] = __builtin_amdgcn_wmma_f32_16x16x4_f32(
                    false, afrag, false, bf1, (short)0, acc1[mt], false, false);
            }
        }

        // D layout: lane&15 = N (vox), VGPR r -> M = r + 8*(lane>>4)
        #pragma unroll
        for (int mt = 0; mt < MTILES; ++mt) {
            int chbase = mt * 16 + lhi * 8;            // channel of VGPR 0
            size_t tens = (mt >= 3) ? IM_OFFSET : 0;   // mt 0..2 re, 3..5 im
            int vbase = (mt >= 3) ? (chbase - V_) : chbase;
            size_t rowb = (size_t)bg * V_ + vbase;
            #pragma unroll
            for (int r = 0; r < 8; ++r) {
                __builtin_nontemporal_store(acc0[mt][r],
                    &out[tens + (rowb + r) * VOL + vox0]);
                __builtin_nontemporal_store(acc1[mt][r],
                    &out[tens + (rowb + r) * VOL + vox1]);
            }
        }
    }
}

// ---- launch -----------------------------------------------------------------
extern "C" void kernel_launch(void* const* d_in, const int* in_sizes, int n_in,
                              void* d_out, int out_size, void* d_ws, size_t ws_size,
                              hipStream_t stream) {
    const float* img   = (const float*)d_in[0];  // batch_images
    const float* cache = (const float*)d_in[1];  // cache
    const float* so3   = (const float*)d_in[2];  // so3_weight
    const float* kr    = (const float*)d_in[3];  // kernel_real
    const float* ki    = (const float*)d_in[4];  // kernel_imag
    float* out = (float*)d_out;

    float* ws    = (float*)d_ws;
    float* apack = ws;                 // 24192 floats (padded region 24576)
    float* Hr    = ws + 24576;         // 688128 floats
    float* Hi    = Hr + H_FLOATS;      // 688128 floats

    const int convLdsBytes = CONV_LDS_FLOATS * 4;   // 161792 B (< 320KB WGP LDS)
    hipFuncSetAttribute((const void*)conv_wmma_kernel,
                        hipFuncAttributeMaxDynamicSharedMemorySize, convLdsBytes);

    prepack_kernel<<<(APACK_FLOATS + 255) / 256, 256, 0, stream>>>(kr, ki, apack);
    stage1_kernel<<<B_ * (VOL / S1_TILEVOX), 256, S1_LDS_FLOATS * 4, stream>>>(
        img, cache, so3, Hr, Hi);
    conv_wmma_kernel<<<B_ * G_ * MSPLIT, 256, convLdsBytes, stream>>>(
        apack, Hr, Hi, out);
}